// BoxFilter_34196529610982
// MI455X (gfx1250) — compile-verified
//
#include <hip/hip_runtime.h>

// Box filter 21x21 (all-ones), circular padding, N=8, C=1, H=W=2048, f32.
// Memory-bound: ~300 MiB HBM traffic -> ~13 us floor at 23.3 TB/s.
//
// Persistent-workgroup fused separable filter with cross-tile double buffering:
//   - prefetch tile i+1 into LDS buffer B via GLOBAL_LOAD_ASYNC_TO_LDS_B64
//     (ASYNCcnt path, no VGPR writeback) while tile i is filtered from buffer A
//   - Stage 2: horizontal 21-tap sliding sums in LDS (VALU, O(1)/pixel)
//   - Stage 3: vertical 21-tap sums as banded-ones matmul via
//     V_WMMA_F32_16X16X4_F32 (exact f32, 9 chained K=4 steps per 16x16 tile)
// LDS: 2 x (148x148) input + 148x136 tmp = ~250 KB of the WGP's 320 KB.

typedef __attribute__((ext_vector_type(2))) float v2f;
typedef __attribute__((ext_vector_type(8))) float v8f;

#define RX 10
#define KW 21               // 2*RX+1
#define IMH 2048
#define IMW 2048
#define TH 128              // output tile height
#define TW 128              // output tile width
#define IN_H (TH + 2*RX)    // 148 rows staged
#define IN_W (TW + 2*RX)    // 148 cols staged
#define IN_STRIDE IN_W      // 148
#define TMP_STRIDE 136      // 2*136 mod 64 == 16 -> lane halves hit disjoint banks
#define NBLK 256            // persistent workgroups (1 per WGP; 250KB LDS each)

__global__ __launch_bounds__(256)
void box21_async_kernel(const float* __restrict__ x, float* __restrict__ out,
                        int n_img) {
    extern __shared__ float smem[];
    float* s_in0 = smem;                            // tile buffer 0
    float* s_in1 = smem + IN_H * IN_STRIDE;         // tile buffer 1
    float* s_tmp = smem + 2 * IN_H * IN_STRIDE;     // horizontal sums

    const int tid = threadIdx.x;
    const int NT  = 16 * 16 * n_img;                // total tiles (2048)

    // ---- async prefetch of one halo tile into an LDS buffer (b64 per lane) ----
    // w0-10+c is always even -> each 2-float pair is wrap-contiguous & 8B aligned.
    auto prefetch = [&](int t, float* dst) {
        int tw = t & 15, th = (t >> 4) & 15, n = t >> 8;
        int w0 = tw * TW, h0 = th * TH;
        const float* img = x + (long long)n * ((long long)IMH * IMW);
        for (int p = tid; p < IN_H * (IN_W / 2); p += 256) {
            int r  = p / (IN_W / 2);
            int c2 = (p - r * (IN_W / 2)) * 2;
            int gh = (h0 + r  - RX) & (IMH - 1);
            int gw = (w0 + c2 - RX) & (IMW - 1);
            unsigned lds = (unsigned)(uintptr_t)&dst[r * IN_STRIDE + c2];
            unsigned long long ga =
                (unsigned long long)(uintptr_t)(img + (long long)gh * IMW + gw);
            asm volatile("global_load_async_to_lds_b64 %0, %1, off"
                         :: "v"(lds), "v"(ga) : "memory");
        }
    };

    // per-lane constants for the WMMA stages
    const int lane = tid & 31;
    const int wv   = tid >> 5;     // 8 waves per block
    const int hl   = lane >> 4;    // lane half (K-offset selector)
    const int lo   = lane & 15;    // M index (A) / N index (B,C,D)

    // A-matrix band fragments: A[m,k]=1 iff m <= k <= m+20, 16x4 f32 layout:
    //   v.x = A[lo, 4*kk + 2*hl], v.y = A[lo, 4*kk + 2*hl + 1]
    v2f a9[9];
    #pragma unroll
    for (int kk = 0; kk < 9; ++kk) {
        int k0 = 4 * kk + 2 * hl;
        a9[kk].x = (k0     >= lo && k0     <= lo + 20) ? 1.f : 0.f;
        a9[kk].y = (k0 + 1 >= lo && k0 + 1 <= lo + 20) ? 1.f : 0.f;
    }

    int t = blockIdx.x;
    if (t < NT) prefetch(t, s_in0);
    int cur = 0;

    for (; t < NT; t += NBLK) {
        // wait for this tile's async fill (own loads), then all waves' fills;
        // this barrier also retires last iteration's s_tmp readers.
        asm volatile("s_wait_asynccnt 0x0" ::: "memory");
        __syncthreads();

        // kick off next tile's fill into the other buffer (overlaps compute)
        int tn = t + NBLK;
        if (tn < NT) prefetch(tn, cur ? s_in0 : s_in1);

        float* s_in = cur ? s_in1 : s_in0;
        int tw = t & 15, th = (t >> 4) & 15, n = t >> 8;
        int w0 = tw * TW, h0 = th * TH;
        const long long obase = (long long)n * ((long long)IMH * IMW);

        // ---- Stage 2: horizontal 21-tap sliding sums (148 rows x 4 chunks) ----
        for (int j = tid; j < IN_H * 4; j += 256) {
            int r  = j >> 2;
            int c0 = (j & 3) * 32;
            const float* row = s_in + r * IN_STRIDE + c0;
            float s = 0.f;
            #pragma unroll
            for (int k = 0; k < KW; ++k) s += row[k];
            float* trow = s_tmp + r * TMP_STRIDE + c0;
            trow[0] = s;
            #pragma unroll
            for (int i = 1; i < 32; ++i) {
                s += row[i + KW - 1] - row[i - 1];
                trow[i] = s;
            }
        }
        __syncthreads();

        // ---- Stage 3: vertical 21-tap sums via WMMA banded matmul ----
        // 64 16x16 subtiles, 8 per wave; uniform loop bounds (EXEC all-1s).
        for (int st = wv; st < (TH / 16) * (TW / 16); st += 8) {
            int m0 = (st >> 3) * 16;
            int c0 = (st & 7) * 16;
            v8f acc = {};
            #pragma unroll
            for (int kk = 0; kk < 9; ++kk) {
                int k0 = 4 * kk + 2 * hl;
                v2f b;  // B 4x16 f32: VGPR0 holds K=2*hl, VGPR1 holds K=2*hl+1
                b.x = s_tmp[(m0 + k0)     * TMP_STRIDE + c0 + lo];
                b.y = s_tmp[(m0 + k0 + 1) * TMP_STRIDE + c0 + lo];
                acc = __builtin_amdgcn_wmma_f32_16x16x4_f32(
                    false, a9[kk], false, b, (short)0, acc, false, false);
            }
            #pragma unroll
            for (int g = 0; g < 8; ++g) {   // C/D: VGPR g -> row g+8*hl, col lo
                int row = h0 + m0 + g + 8 * hl;
                int col = w0 + c0 + lo;
                out[obase + (long long)row * IMW + col] = acc[g];
            }
        }
        cur ^= 1;
    }
}

extern "C" void kernel_launch(void* const* d_in, const int* in_sizes, int n_in,
                              void* d_out, int out_size, void* d_ws, size_t ws_size,
                              hipStream_t stream) {
    const float* x = (const float*)d_in[0];
    // d_in[1] is the all-ones 21x21 kernel; the box sum is baked in.
    float* out = (float*)d_out;

    int n_img = in_sizes[0] / (IMH * IMW);   // 8
    int nt = 16 * 16 * n_img;
    int nblk = nt < NBLK ? nt : NBLK;
    size_t shmem = (size_t)(2 * IN_H * IN_STRIDE + IN_H * TMP_STRIDE) * sizeof(float);
    box21_async_kernel<<<dim3(nblk), 256, shmem, stream>>>(x, out, n_img);
}